// Extractor_87651692577363
// MI455X (gfx1250) — compile-verified
//
#include <hip/hip_runtime.h>
#include <math.h>

// Problem geometry (fixed by the reference harness)
#define HH    240
#define WW    320
#define NPIX  (HH * WW)          // 76800
#define NRAY  9
#define NSAMP (NPIX * NRAY)      // 691200
#define VD    256                // volume dim

// d_out float offsets, in reference return order
#define OFF_FV  0                               // fusion_values  [NSAMP]
#define OFF_FW  (NSAMP)                         // fusion_weights [NSAMP]
#define OFF_RP  (2 * NSAMP)                     // ray_pts        [NSAMP*3]
#define OFF_DF  (5 * NSAMP)                     // depth_flat     [NPIX]
#define OFF_IDX (5 * NSAMP + NPIX)              // indices        [NSAMP*8*3]
#define OFF_W8  (OFF_IDX + NSAMP * 24)          // weights8       [NSAMP*8]
#define OFF_CO  (OFF_W8 + NSAMP * 8)            // coords         [NPIX*3]

typedef float v2f __attribute__((ext_vector_type(2)));
typedef float v8f __attribute__((ext_vector_type(8)));

__device__ __forceinline__ float sgnf(float x) {
    return (x > 0.0f) ? 1.0f : ((x < 0.0f) ? -1.0f : 0.0f);
}

// ---------------------------------------------------------------------------
// Kernel 1: world-coordinate transform as a (4x4) x (4xN) FP32 GEMM using
// V_WMMA_F32_16X16X4_F32. Each wave32 transforms 16 pixels per WMMA.
//   A (16x4): combined matrix M = E * [[Kinv,0],[0,1]] in rows 0..3, rest 0
//   B (4x16): 16 homogeneous points [jj*z, ii*z, z, 1] as columns
//   D rows 0..2 (VGPRs 0..2, lanes 0..15) = world x,y,z of point N=lane
// No dynamically-indexed private arrays (avoids LDS/scratch lowering);
// no exec-mask manipulation before the WMMA (grid covers NPIX exactly).
// ---------------------------------------------------------------------------
__global__ void coords_wmma_kernel(const float* __restrict__ depth,
                                   const float* __restrict__ intr,
                                   const float* __restrict__ extr,
                                   float* __restrict__ out) {
    const int lane = threadIdx.x & 31;
    const int wave = (blockIdx.x * blockDim.x + threadIdx.x) >> 5;
    const int base = wave * 16;          // 4800 waves * 16 == NPIX exactly

    // 3x3 inverse of intrinsics (adjugate / det) as individual scalars
    const float a = intr[0], b = intr[1], c = intr[2];
    const float d = intr[3], e = intr[4], f = intr[5];
    const float g = intr[6], h = intr[7], ia = intr[8];
    const float A0 = e * ia - f * h;
    const float A1 = -(d * ia - f * g);
    const float A2 = d * h - e * g;
    const float rdet = 1.0f / (a * A0 + b * A1 + c * A2);
    const float Ki00 = A0 * rdet, Ki01 = -(b * ia - c * h) * rdet, Ki02 = (b * f - c * e) * rdet;
    const float Ki10 = A1 * rdet, Ki11 =  (a * ia - c * g) * rdet, Ki12 = -(a * f - c * d) * rdet;
    const float Ki20 = A2 * rdet, Ki21 = -(a * h - b * g) * rdet,  Ki22 = (a * e - b * d) * rdet;

    const int  row = lane & 15;
    const bool hi  = lane >= 16;
    const int  r4  = row & 3;            // clamp row for in-bounds extrinsics read

    // This lane's extrinsics row (rows >= 4 of A are zeroed below anyway)
    const float e0 = extr[r4 * 4 + 0];
    const float e1 = extr[r4 * 4 + 1];
    const float e2 = extr[r4 * 4 + 2];
    const float e3 = extr[r4 * 4 + 3];

    // A operand: 16x4 fp32. VGPR0 holds K = (hi?2:0), VGPR1 holds K = (hi?3:1).
    // M[r][c<3] = E[r][0..2] . Ki[:,c] ;  M[r][3] = E[r][3]
    const float kA0 = hi ? Ki02 : Ki00;
    const float kA1 = hi ? Ki12 : Ki10;
    const float kA2 = hi ? Ki22 : Ki20;
    const float mA  = e0 * kA0 + e1 * kA1 + e2 * kA2;                   // col 0 or 2
    const float mB  = hi ? e3 : (e0 * Ki01 + e1 * Ki11 + e2 * Ki21);    // col 1 or 3
    v2f Av;
    Av.x = (row < 4) ? mA : 0.0f;
    Av.y = (row < 4) ? mB : 0.0f;

    // B operand: 4x16 fp32, column N = lane&15 is the homogeneous pixel point
    const int   pt  = base + row;
    const float z   = depth[pt];
    const float iif = (float)(pt / WW);
    const float jjf = (float)(pt % WW);
    v2f Bv;
    Bv.x = hi ? z    : jjf * z;   // K = 2 : 0   (yy*z)
    Bv.y = hi ? 1.0f : iif * z;   // K = 3 : 1   (xx*z)

    v8f Cv = {};
    v8f Dv = __builtin_amdgcn_wmma_f32_16x16x4_f32(
        /*neg_a=*/false, Av, /*neg_b=*/false, Bv,
        /*c_mod=*/(short)0, Cv, /*reuse_a=*/false, /*reuse_b=*/false);

    // Rows 0..2 (world x,y,z) live in VGPRs 0..2 on lanes 0..15
    if (!hi) {
        out[OFF_CO + pt * 3 + 0] = Dv[0];
        out[OFF_CO + pt * 3 + 1] = Dv[1];
        out[OFF_CO + pt * 3 + 2] = Dv[2];
    }
}

// ---------------------------------------------------------------------------
// Kernel 2: depth_flat is a verbatim dense 300 KB copy -> use the CDNA5
// async Global<->LDS engine (ASYNCcnt-tracked). The builtins expect
// addrspace(1) int* (global) / addrspace(3) int* (LDS) parameters.
// ---------------------------------------------------------------------------
#if __has_builtin(__builtin_amdgcn_global_load_async_to_lds_b32) && \
    __has_builtin(__builtin_amdgcn_global_store_async_from_lds_b32)
#define USE_ASYNC_LDS_COPY 1
#else
#define USE_ASYNC_LDS_COPY 0
#endif

__global__ void depth_copy_kernel(const float* __restrict__ depth,
                                  float* __restrict__ out) {
    const int i = blockIdx.x * blockDim.x + threadIdx.x;   // 300*256 == NPIX exactly
#if USE_ASYNC_LDS_COPY
    __shared__ int lbuf[256];
    typedef __attribute__((address_space(1))) int* g_i32p;
    typedef __attribute__((address_space(3))) int* lds_i32p;
    lds_i32p lp = (lds_i32p)&lbuf[threadIdx.x];
    // global -> LDS (ASYNCcnt++)
    __builtin_amdgcn_global_load_async_to_lds_b32((g_i32p)(depth + i), lp, 0, 0);
    asm volatile("s_wait_asynccnt 0x0" ::: "memory");
    // LDS -> global (ASYNCcnt++)
    __builtin_amdgcn_global_store_async_from_lds_b32((g_i32p)(out + OFF_DF + i), lp, 0, 0);
    asm volatile("s_wait_asynccnt 0x0" ::: "memory");
#else
    out[OFF_DF + i] = depth[i];
#endif
}

// ---------------------------------------------------------------------------
// Kernel 3: one thread per (pixel, ray) sample. Reads coords written by the
// WMMA kernel, walks the ray, performs the 8-corner trilinear gather from the
// two 256^3 volumes (L2-resident: 128 MB < 192 MB L2), streams all outputs.
// indices / weights8 are staged in registers and stored as float4 -> b128.
// ---------------------------------------------------------------------------
__global__ void extract_kernel(const float* __restrict__ extr,
                               const float* __restrict__ tsdf,
                               const float* __restrict__ wvol,
                               const float* __restrict__ origin,
                               const float* __restrict__ resolution,
                               float* out) {
    const int t = blockIdx.x * blockDim.x + threadIdx.x;
    if (t >= NSAMP) return;
    const int   p    = t / NRAY;
    const int   k    = t - p * NRAY;
    const float offs = (float)(k - 4);          // -4..4

    const float rres = 1.0f / resolution[0];
    const float ox = origin[0], oy = origin[1], oz = origin[2];

    // voxel-space center point and eye
    const float cvx = (out[OFF_CO + p * 3 + 0] - ox) * rres;
    const float cvy = (out[OFF_CO + p * 3 + 1] - oy) * rres;
    const float cvz = (out[OFF_CO + p * 3 + 2] - oz) * rres;
    const float evx = (extr[3]  - ox) * rres;
    const float evy = (extr[7]  - oy) * rres;
    const float evz = (extr[11] - oz) * rres;

    float dx = cvx - evx, dy = cvy - evy, dz = cvz - evz;
    const float nrm  = sqrtf(dx * dx + dy * dy + dz * dz);
    const float rinv = 1.0f / fmaxf(nrm, 1e-12f);
    dx *= rinv; dy *= rinv; dz *= rinv;

    const float Px = cvx + offs * dx;
    const float Py = cvy + offs * dy;
    const float Pz = cvz + offs * dz;
    out[OFF_RP + t * 3 + 0] = Px;
    out[OFF_RP + t * 3 + 1] = Py;
    out[OFF_RP + t * 3 + 2] = Pz;

    // trilinear setup (reference's floor+0.5 / sign / abs scheme)
    const float fx = floorf(Px), fy = floorf(Py), fz = floorf(Pz);
    const float ctx = fx + 0.5f, cty = fy + 0.5f, ctz = fz + 0.5f;
    const float nx = sgnf(ctx - Px), ny = sgnf(cty - Py), nz = sgnf(ctz - Pz);
    const float ax = fabsf(Px - ctx), ay = fabsf(Py - cty), az = fabsf(Pz - ctz);
    const float axi = 1.0f - ax, ayi = 1.0f - ay, azi = 1.0f - az;

    float idxb[24];   // constant-indexed after unroll -> stays in VGPRs
    float w8b[8];
    float fv = 0.0f, fw = 0.0f;
    #pragma unroll
    for (int c = 0; c < 8; ++c) {
        const int bi = (c >> 2) & 1, bj = (c >> 1) & 1, bk = c & 1;
        const float w3 = (bi ? ax : axi) * (bj ? ay : ayi) * (bk ? az : azi);

        const float c0f = fx + (bi ? nx : 0.0f);
        const float c1f = fy + (bj ? ny : 0.0f);
        const float c2f = fz + (bk ? nz : 0.0f);
        const int c0 = (int)c0f, c1 = (int)c1f, c2 = (int)c2f;

        const bool valid = (c0 >= 0) && (c0 < VD) &&
                           (c1 >= 0) && (c1 < VD) &&
                           (c2 >= 0) && (c2 < VD);
        const int q0 = min(max(c0, 0), VD - 1);
        const int q1 = min(max(c1, 0), VD - 1);
        const int q2 = min(max(c2, 0), VD - 1);
        const int lin = (q0 * VD + q1) * VD + q2;

        const float vm = valid ? 1.0f : 0.0f;
        fv += tsdf[lin] * vm * w3;
        fw += wvol[lin] * vm * w3;

        idxb[c * 3 + 0] = (float)c0;
        idxb[c * 3 + 1] = (float)c1;
        idxb[c * 3 + 2] = (float)c2;
        w8b[c] = w3;
    }

    // 24 floats (96 B) of indices as 6 x b128 (16 B aligned: OFF_IDX*4 % 16 == 0)
    float4* ip = reinterpret_cast<float4*>(out + OFF_IDX + t * 24);
    #pragma unroll
    for (int q = 0; q < 6; ++q)
        ip[q] = make_float4(idxb[4 * q + 0], idxb[4 * q + 1],
                            idxb[4 * q + 2], idxb[4 * q + 3]);

    // 8 floats (32 B) of weights as 2 x b128
    float4* wp = reinterpret_cast<float4*>(out + OFF_W8 + t * 8);
    wp[0] = make_float4(w8b[0], w8b[1], w8b[2], w8b[3]);
    wp[1] = make_float4(w8b[4], w8b[5], w8b[6], w8b[7]);

    out[OFF_FV + t] = fv;
    out[OFF_FW + t] = fw;
}

extern "C" void kernel_launch(void* const* d_in, const int* in_sizes, int n_in,
                              void* d_out, int out_size, void* d_ws, size_t ws_size,
                              hipStream_t stream) {
    (void)in_sizes; (void)n_in; (void)out_size; (void)d_ws; (void)ws_size;
    const float* depth = (const float*)d_in[0];
    const float* extr  = (const float*)d_in[1];
    const float* intr  = (const float*)d_in[2];
    const float* tsdf  = (const float*)d_in[3];
    const float* orig  = (const float*)d_in[4];
    const float* reso  = (const float*)d_in[5];
    const float* wvol  = (const float*)d_in[6];
    float* out = (float*)d_out;

    // 4800 waves of 16 pixels each, 8 waves (256 threads) per block -> 600 blocks
    coords_wmma_kernel<<<600, 256, 0, stream>>>(depth, intr, extr, out);
    // dense depth -> depth_flat copy via async LDS engine
    depth_copy_kernel<<<NPIX / 256, 256, 0, stream>>>(depth, out);
    // one thread per (pixel, ray) sample
    extract_kernel<<<(NSAMP + 255) / 256, 256, 0, stream>>>(
        extr, tsdf, wvol, orig, reso, out);
}